// QCNN_64347200028752
// MI455X (gfx1250) — compile-verified
//
#include <hip/hip_runtime.h>

// ---------------------------------------------------------------------------
// QCNN: conv(3x3,s2) as fp32 WMMA GEMM  +  per-thread 4-qubit statevector sim
// Target: gfx1250 (CDNA5, wave32, V_WMMA_F32_16X16X4_F32)
//
// v2: B (weight) matrix built ONCE per block into LDS table; K-loop reads it
//     with a single ds_load_b64 per chunk -> removes the per-chunk
//     v_cmp/v_cndmask select chains and saveexec blocks seen in v1.
// ---------------------------------------------------------------------------

typedef __attribute__((ext_vector_type(2))) float v2f;
typedef __attribute__((ext_vector_type(8))) float v8f;

#define BATCH 262144
#define IMG   63      // 7*9 image pixels == GEMM K (padded to 64)
#define NPOS  12      // 3*4 conv outputs == GEMM N (padded to 16)
#define BLK   256
#define BPAD  66      // B-table row pad: keeps m*BPAD+k 8-byte aligned, spreads banks

// ---- quantum gate helpers (stride is a template const -> indices fold) ----
template <int STRIDE>
__device__ __forceinline__ void rx_gate(float (&sr)[16], float (&si)[16],
                                        float c, float s) {
#pragma unroll
  for (int i0 = 0; i0 < 16; ++i0) {
    if (i0 & STRIDE) continue;
    const int i1 = i0 | STRIDE;
    float a0r = sr[i0], a0i = si[i0], a1r = sr[i1], a1i = si[i1];
    sr[i0] = c * a0r + s * a1i;  si[i0] = c * a0i - s * a1r;
    sr[i1] = c * a1r + s * a0i;  si[i1] = c * a1i - s * a0r;
  }
}

template <int STRIDE>
__device__ __forceinline__ void ry_gate(float (&sr)[16], float (&si)[16],
                                        float c, float s) {
#pragma unroll
  for (int i0 = 0; i0 < 16; ++i0) {
    if (i0 & STRIDE) continue;
    const int i1 = i0 | STRIDE;
    float a0r = sr[i0], a0i = si[i0], a1r = sr[i1], a1i = si[i1];
    sr[i0] = c * a0r - s * a1r;  si[i0] = c * a0i - s * a1i;
    sr[i1] = s * a0r + c * a1r;  si[i1] = s * a0i + c * a1i;
  }
}

// ISWAP on qubits a (stride SA) and b (stride SB): |01> -> i|10>, |10> -> i|01>
template <int SA, int SB>
__device__ __forceinline__ void iswap_gate(float (&sr)[16], float (&si)[16]) {
#pragma unroll
  for (int i = 0; i < 16; ++i) {
    if (i & (SA | SB)) continue;
    const int i01 = i | SB;  // a=0,b=1
    const int i10 = i | SA;  // a=1,b=0
    float xr = sr[i01], xi = si[i01];
    float yr = sr[i10], yi = si[i10];
    sr[i01] = -yi; si[i01] = yr;   // i * old(a=1,b=0)
    sr[i10] = -xi; si[i10] = xr;   // i * old(a=0,b=1)
  }
}

// weight-matrix entry W[t][p] for tap t (0..63), output position p (0..11)
__device__ __forceinline__ float wmat(const float (&cw)[9], int t, int p) {
  if (t >= IMG) return 0.0f;
  const int iy = t / 9, ix = t - iy * 9;
  const int oy = p >> 2, ox = p & 3;
  const int ky = iy - 2 * oy, kx = ix - 2 * ox;
  if (ky >= 0 && ky < 3 && kx >= 0 && kx < 3) return cw[ky * 3 + kx];
  return 0.0f;
}

__global__ __launch_bounds__(BLK) void qcnn_main(
    const float* __restrict__ images, const int* __restrict__ y,
    const float* __restrict__ conv_w, const float* __restrict__ conv_b,
    const float* __restrict__ qp, float* __restrict__ block_sums) {
  __shared__ __align__(16) float s_B[16][BPAD];  // B^T: [n][k], zero-padded
  __shared__ float s_conv[BLK][NPOS];
  __shared__ float s_red[BLK];

  const int tid  = threadIdx.x;
  const int lane = tid & 31;
  const int wv   = tid >> 5;
  const int m    = lane & 15;  // A row (M) / B,D column (N)
  const int h    = lane >> 4;  // lane half -> K split

  // ---- build the 16(N) x 64(K) weight table once per block ----------------
  {
    float cw[9];
#pragma unroll
    for (int i = 0; i < 9; ++i) cw[i] = conv_w[i];
    const int n  = tid >> 4;          // 0..15
    const int k0 = (tid & 15) * 4;    // 0,4,...,60
#pragma unroll
    for (int j = 0; j < 4; ++j)
      s_B[n][k0 + j] = (n < NPOS) ? wmat(cw, k0 + j, n) : 0.0f;
  }
  __syncthreads();

  // ---- conv as WMMA: 2 tiles of 16 batch rows per wave, K = 16 chunks of 4
  const int wave_batch0 = blockIdx.x * BLK + wv * 32;
  const float* row0 = images + (size_t)(wave_batch0 + m) * IMG;
  const float* row1 = images + (size_t)(wave_batch0 + 16 + m) * IMG;

  v8f c0 = {};
  v8f c1 = {};

#pragma unroll
  for (int k4 = 0; k4 < 16; ++k4) {
    const int k = k4 * 4 + h * 2;  // K of reg0 element (reg1 = k+1), even

    // B fragment: one aligned 8-byte LDS read
    const v2f bfrag = *(const v2f*)(&s_B[m][k]);

    v2f a0, a1;
    a0.x = row0[k];
    a0.y = (k + 1 < IMG) ? row0[k + 1] : 0.0f;   // folds for k4 < 15
    a1.x = row1[k];
    a1.y = (k + 1 < IMG) ? row1[k + 1] : 0.0f;

    c0 = __builtin_amdgcn_wmma_f32_16x16x4_f32(false, a0, false, bfrag,
                                               (short)0, c0, false, false);
    c1 = __builtin_amdgcn_wmma_f32_16x16x4_f32(false, a1, false, bfrag,
                                               (short)0, c1, false, false);
  }

  // D layout: lane holds N = lane%16; VGPR r holds M = r + 8*(lane/16)
  if (m < NPOS) {
#pragma unroll
    for (int r = 0; r < 8; ++r) {
      s_conv[wv * 32 + (r + 8 * h)][m]      = c0[r];
      s_conv[wv * 32 + 16 + (r + 8 * h)][m] = c1[r];
    }
  }
  __syncthreads();

  // ---- per-thread 4-qubit statevector simulation -------------------------
  const float bias = conv_b[0];
  float ac[12], as[12];
#pragma unroll
  for (int i = 0; i < 12; ++i) {
    const float v = s_conv[tid][i] + bias;
    float sv, cv;
    sincosf(0.5f * v, &sv, &cv);
    ac[i] = cv; as[i] = sv;
  }

  float sr[16], si[16];
#pragma unroll
  for (int i = 0; i < 16; ++i) { sr[i] = 0.0f; si[i] = 0.0f; }
  sr[0] = 1.0f;

  // layer 1: per wire w: RX(d0) RY(d1) RX(d2); wire w -> stride 8>>w
  rx_gate<8>(sr, si, ac[0],  as[0]);
  ry_gate<8>(sr, si, ac[1],  as[1]);
  rx_gate<8>(sr, si, ac[2],  as[2]);
  rx_gate<4>(sr, si, ac[3],  as[3]);
  ry_gate<4>(sr, si, ac[4],  as[4]);
  rx_gate<4>(sr, si, ac[5],  as[5]);
  rx_gate<2>(sr, si, ac[6],  as[6]);
  ry_gate<2>(sr, si, ac[7],  as[7]);
  rx_gate<2>(sr, si, ac[8],  as[8]);
  rx_gate<1>(sr, si, ac[9],  as[9]);
  ry_gate<1>(sr, si, ac[10], as[10]);
  rx_gate<1>(sr, si, ac[11], as[11]);

  // ISWAP (a,b) = (1,0),(2,1),(3,2) -> strides (4,8),(2,4),(1,2)
  iswap_gate<4, 8>(sr, si);
  iswap_gate<2, 4>(sr, si);
  iswap_gate<1, 2>(sr, si);

  // layer 2: same angles, RY(d0) RX(d1) RY(d2)
  ry_gate<8>(sr, si, ac[0],  as[0]);
  rx_gate<8>(sr, si, ac[1],  as[1]);
  ry_gate<8>(sr, si, ac[2],  as[2]);
  ry_gate<4>(sr, si, ac[3],  as[3]);
  rx_gate<4>(sr, si, ac[4],  as[4]);
  ry_gate<4>(sr, si, ac[5],  as[5]);
  ry_gate<2>(sr, si, ac[6],  as[6]);
  rx_gate<2>(sr, si, ac[7],  as[7]);
  ry_gate<2>(sr, si, ac[8],  as[8]);
  ry_gate<1>(sr, si, ac[9],  as[9]);
  rx_gate<1>(sr, si, ac[10], as[10]);
  ry_gate<1>(sr, si, ac[11], as[11]);

  // ISWAP (3,2),(2,1),(1,0)
  iswap_gate<1, 2>(sr, si);
  iswap_gate<2, 4>(sr, si);
  iswap_gate<4, 8>(sr, si);

  // shared RX,RY,RX on wire 0
  {
    float s0, c0q, s1, c1q, s2, c2q;
    sincosf(0.5f * qp[0], &s0, &c0q);
    sincosf(0.5f * qp[1], &s1, &c1q);
    sincosf(0.5f * qp[2], &s2, &c2q);
    rx_gate<8>(sr, si, c0q, s0);
    ry_gate<8>(sr, si, c1q, s1);
    rx_gate<8>(sr, si, c2q, s2);
  }

  // p1 = sum over q0==1 amplitudes (indices 8..15)
  float p1 = 0.0f;
#pragma unroll
  for (int i = 8; i < 16; ++i) p1 += sr[i] * sr[i] + si[i] * si[i];

  const int gb = blockIdx.x * BLK + tid;
  const float expv = y[gb] ? p1 : (1.0f - p1);
  const float d = 1.0f - expv;
  s_red[tid] = d * d;
  __syncthreads();

#pragma unroll
  for (int off = BLK / 2; off > 0; off >>= 1) {
    if (tid < off) s_red[tid] += s_red[tid + off];
    __syncthreads();
  }
  if (tid == 0) block_sums[blockIdx.x] = s_red[0];
}

__global__ __launch_bounds__(256) void qcnn_reduce(
    const float* __restrict__ bs, float* __restrict__ out, int n, float invB) {
  __shared__ float red[256];
  float acc = 0.0f;
  for (int i = threadIdx.x; i < n; i += 256) acc += bs[i];
  red[threadIdx.x] = acc;
  __syncthreads();
#pragma unroll
  for (int off = 128; off > 0; off >>= 1) {
    if (threadIdx.x < off) red[threadIdx.x] += red[threadIdx.x + off];
    __syncthreads();
  }
  if (threadIdx.x == 0) out[0] = red[0] * invB;
}

extern "C" void kernel_launch(void* const* d_in, const int* in_sizes, int n_in,
                              void* d_out, int out_size, void* d_ws,
                              size_t ws_size, hipStream_t stream) {
  const float* images = (const float*)d_in[0];
  const int*   y      = (const int*)d_in[1];
  const float* conv_w = (const float*)d_in[2];
  const float* conv_b = (const float*)d_in[3];
  const float* qp     = (const float*)d_in[4];
  float* out   = (float*)d_out;
  float* bsums = (float*)d_ws;  // 1024 floats of scratch

  const int nblocks = BATCH / BLK;  // 1024
  qcnn_main<<<nblocks, BLK, 0, stream>>>(images, y, conv_w, conv_b, qp, bsums);
  qcnn_reduce<<<1, 256, 0, stream>>>(bsums, out, nblocks, 1.0f / (float)BATCH);
}